// PoneGNNEncoder_89739046682757
// MI455X (gfx1250) — compile-verified
//
#include <hip/hip_runtime.h>
#include <hip/hip_bf16.h>
#include <stdint.h>

namespace {
constexpr int   kNumUsers = 100000;
constexpr int   kN        = 300000;          // users + items
constexpr int   kD        = 64;
constexpr int   kVec      = kD / 4;          // 16 float4 per row
constexpr float kAlpha    = 1.0f / 3.0f;     // 1/(NUM_LAYERS+1)
constexpr int   kBlock    = 256;             // 8 wave32 waves
}

// ---------------------------------------------------------------------------
// Fused pos‖neg: ego = concat(user, item); out = alpha * ego
// Blocks [0, half) -> pos path, [half, 2*half) -> neg path.
// ---------------------------------------------------------------------------
__global__ __launch_bounds__(kBlock)
void build_ego2_kernel(const float4* __restrict__ user_p,
                       const float4* __restrict__ item_p,
                       const float4* __restrict__ user_n,
                       const float4* __restrict__ item_n,
                       float4* __restrict__ ego_p, float4* __restrict__ ego_n,
                       float4* __restrict__ out_p, float4* __restrict__ out_n,
                       int half_blocks) {
  const float4* user_emb; const float4* item_emb; float4* ego; float4* out;
  int blk = blockIdx.x;
  if (blk < half_blocks) {
    user_emb = user_p; item_emb = item_p; ego = ego_p; out = out_p;
  } else {
    user_emb = user_n; item_emb = item_n; ego = ego_n; out = out_n;
    blk -= half_blocks;
  }
  long long i = (long long)blk * blockDim.x + threadIdx.x;
  if (i >= (long long)kN * kVec) return;
  long long node = i >> 4;
  float4 v = (node < kNumUsers) ? user_emb[i]
                                : item_emb[i - (long long)kNumUsers * kVec];
  ego[i] = v;
  float4 o;
  o.x = v.x * kAlpha; o.y = v.y * kAlpha; o.z = v.z * kAlpha; o.w = v.w * kAlpha;
  out[i] = o;
}

// ---------------------------------------------------------------------------
// Fused pos‖neg degree count. Edge indices staged global->LDS via the CDNA5
// async path (ASYNCcnt); each lane consumes its own slot after the wait.
// ---------------------------------------------------------------------------
__global__ __launch_bounds__(kBlock)
void degree2_kernel(const int* __restrict__ col_p, long long e_p,
                    float* __restrict__ deg_p,
                    const int* __restrict__ col_n, long long e_n,
                    float* __restrict__ deg_n, int pos_blocks) {
  __shared__ int s_col[kBlock];
  const int* col; long long E; float* deg;
  int blk = blockIdx.x;
  if (blk < pos_blocks) { col = col_p; E = e_p; deg = deg_p; }
  else                  { col = col_n; E = e_n; deg = deg_n; blk -= pos_blocks; }
  int tid = threadIdx.x;
  long long idx = (long long)blk * blockDim.x + tid;
  if (idx < E) {
    unsigned lds_off = (unsigned)(uintptr_t)&s_col[tid];   // low 32b == LDS offset
    unsigned long long gaddr = (unsigned long long)(uintptr_t)(col + idx);
    asm volatile("global_load_async_to_lds_b32 %0, %1, off"
                 :: "v"(lds_off), "v"(gaddr) : "memory");
    asm volatile("s_wait_asynccnt 0" ::: "memory");
    atomicAdd(&deg[s_col[tid]], 1.0f);
  }
}

// deg -> deg^-1/2 in place (0 where deg==0)
__global__ __launch_bounds__(kBlock)
void deg_to_dis_kernel(float* __restrict__ deg, int n) {
  int i = blockIdx.x * blockDim.x + threadIdx.x;
  if (i < n) {
    float d = deg[i];
    deg[i] = (d > 0.0f) ? rsqrtf(d) : 0.0f;
  }
}

// ---------------------------------------------------------------------------
// Layer 0, fused across graphs: out[col] += dis[row]*dis[col]*x[row].
// 16 lanes per edge, float4 each -> 256B coalesced gather + scatter.
// Atomics land in L2 (accumulator buffer fits in 192MB L2).
// ---------------------------------------------------------------------------
__global__ __launch_bounds__(kBlock)
void scatter_l0_kernel(const int* __restrict__ row_p, const int* __restrict__ col_p,
                       long long e_p, const float4* __restrict__ x_p,
                       const float* __restrict__ dis_p, float* __restrict__ out_p,
                       const int* __restrict__ row_n, const int* __restrict__ col_n,
                       long long e_n, const float4* __restrict__ x_n,
                       const float* __restrict__ dis_n, float* __restrict__ out_n,
                       int pos_blocks) {
  const int* row; const int* col; long long E;
  const float4* x; const float* dis; float* out;
  int blk = blockIdx.x;
  if (blk < pos_blocks) {
    row = row_p; col = col_p; E = e_p; x = x_p; dis = dis_p; out = out_p;
  } else {
    row = row_n; col = col_n; E = e_n; x = x_n; dis = dis_n; out = out_n;
    blk -= pos_blocks;
  }
  long long t = (long long)blk * blockDim.x + threadIdx.x;
  long long e = t >> 4;
  if (e >= E) return;
  int d4 = (int)(t & 15);
  __builtin_prefetch(row + e + 4096, 0, 0);   // global_prefetch_b8, speculative
  __builtin_prefetch(col + e + 4096, 0, 0);
  int r = row[e];
  int c = col[e];
  float nrm = dis[r] * dis[c];
  if (nrm != 0.0f) {
    float4 v = x[(long long)r * kVec + d4];
    float* o = out + (long long)c * kD + d4 * 4;
    atomicAdd(o + 0, nrm * v.x);
    atomicAdd(o + 1, nrm * v.y);
    atomicAdd(o + 2, nrm * v.z);
    atomicAdd(o + 3, nrm * v.w);
  }
}

// Layer 1: both paths share the pos graph — read indices/norm once per edge.
__global__ __launch_bounds__(kBlock)
void scatter2_kernel(const int* __restrict__ row, const int* __restrict__ col,
                     long long E,
                     const float4* __restrict__ xp, const float4* __restrict__ xn,
                     const float* __restrict__ dis,
                     float* __restrict__ outp, float* __restrict__ outn) {
  long long t = (long long)blockIdx.x * blockDim.x + threadIdx.x;
  long long e = t >> 4;
  if (e >= E) return;
  int d4 = (int)(t & 15);
  __builtin_prefetch(row + e + 4096, 0, 0);
  __builtin_prefetch(col + e + 4096, 0, 0);
  int r = row[e];
  int c = col[e];
  float nrm = dis[r] * dis[c];
  if (nrm != 0.0f) {
    long long src = (long long)r * kVec + d4;
    long long dst = (long long)c * kD + d4 * 4;
    float4 vp = xp[src];
    float* op = outp + dst;
    atomicAdd(op + 0, nrm * vp.x);
    atomicAdd(op + 1, nrm * vp.y);
    atomicAdd(op + 2, nrm * vp.z);
    atomicAdd(op + 3, nrm * vp.w);
    float4 vn = xn[src];
    float* on = outn + dst;
    atomicAdd(on + 0, nrm * vn.x);
    atomicAdd(on + 1, nrm * vn.y);
    atomicAdd(on + 2, nrm * vn.z);
    atomicAdd(on + 3, nrm * vn.w);
  }
}

// ---------------------------------------------------------------------------
// Fused pos‖neg GIN + layer accumulation:
//   v = prop + (1+eps)*dis^2*ego ; ego <- v ; out += alpha*v
// ---------------------------------------------------------------------------
__global__ __launch_bounds__(kBlock)
void gin_accum2_kernel(const float4* __restrict__ prop_p,
                       const float4* __restrict__ prop_n,
                       float4* __restrict__ ego_p, float4* __restrict__ ego_n,
                       const float* __restrict__ dis_p,
                       const float* __restrict__ dis_n,
                       const float* __restrict__ eps, int layer,
                       float4* __restrict__ out_p, float4* __restrict__ out_n,
                       int half_blocks) {
  const float4* prop; float4* ego; const float* dis; float4* out;
  int blk = blockIdx.x;
  if (blk < half_blocks) { prop = prop_p; ego = ego_p; dis = dis_p; out = out_p; }
  else { prop = prop_n; ego = ego_n; dis = dis_n; out = out_n; blk -= half_blocks; }
  long long i = (long long)blk * blockDim.x + threadIdx.x;
  if (i >= (long long)kN * kVec) return;
  long long node = i >> 4;
  float s = dis[node];
  float k = (1.0f + eps[layer]) * s * s;
  float4 p = prop[i];
  float4 x = ego[i];
  float4 v;
  v.x = fmaf(k, x.x, p.x);
  v.y = fmaf(k, x.y, p.y);
  v.z = fmaf(k, x.z, p.z);
  v.w = fmaf(k, x.w, p.w);
  ego[i] = v;
  float4 o = out[i];
  o.x = fmaf(kAlpha, v.x, o.x);
  o.y = fmaf(kAlpha, v.y, o.y);
  o.z = fmaf(kAlpha, v.z, o.z);
  o.w = fmaf(kAlpha, v.w, o.w);
  out[i] = o;
}

extern "C" void kernel_launch(void* const* d_in, const int* in_sizes, int n_in,
                              void* d_out, int out_size, void* d_ws, size_t ws_size,
                              hipStream_t stream) {
  (void)n_in; (void)out_size; (void)ws_size;
  const int*   pos_ei   = (const int*)d_in[0];     // [2, E_pos] rows then cols
  const int*   neg_ei   = (const int*)d_in[1];     // [2, E_neg]
  const float* user_emb = (const float*)d_in[2];
  const float* item_emb = (const float*)d_in[3];
  const float* user_neg = (const float*)d_in[4];
  const float* item_neg = (const float*)d_in[5];
  const float* eps      = (const float*)d_in[6];
  float* out = (float*)d_out;                       // pos_out | neg_out

  long long e_pos = in_sizes[0] / 2;
  long long e_neg = in_sizes[1] / 2;
  const int* pos_row = pos_ei;
  const int* pos_col = pos_ei + e_pos;
  const int* neg_row = neg_ei;
  const int* neg_col = neg_ei + e_neg;

  size_t per = (size_t)kN * kD;
  float* ws      = (float*)d_ws;
  float* ego_pos = ws;
  float* ego_neg = ws + per;
  float* bop     = ws + 2 * per;
  float* bon     = ws + 3 * per;
  float* dis_pos = ws + 4 * per;
  float* dis_neg = dis_pos + kN;

  const int B = kBlock;
  long long nvec = (long long)kN * kVec;
  int gVec     = (int)((nvec + B - 1) / B);
  int gScatPos = (int)((e_pos * 16 + B - 1) / B);
  int gScatNeg = (int)((e_neg * 16 + B - 1) / B);
  int gDegPos  = (int)((e_pos + B - 1) / B);
  int gDegNeg  = (int)((e_neg + B - 1) / B);

  // ego + alpha-scaled layer-0 contribution (pos ‖ neg in one launch)
  build_ego2_kernel<<<2 * gVec, B, 0, stream>>>(
      (const float4*)user_emb, (const float4*)item_emb,
      (const float4*)user_neg, (const float4*)item_neg,
      (float4*)ego_pos, (float4*)ego_neg,
      (float4*)out, (float4*)(out + per), gVec);

  // degrees -> dis (in place), pos ‖ neg fused
  hipMemsetAsync(dis_pos, 0, 2 * (size_t)kN * sizeof(float), stream);
  degree2_kernel<<<gDegPos + gDegNeg, B, 0, stream>>>(
      pos_col, e_pos, dis_pos, neg_col, e_neg, dis_neg, gDegPos);
  deg_to_dis_kernel<<<(2 * kN + B - 1) / B, B, 0, stream>>>(dis_pos, 2 * kN);

  // ---- Layer 0: pos path on pos graph ‖ neg path on neg graph ----
  hipMemsetAsync(bop, 0, 2 * per * sizeof(float), stream);   // covers bop+bon
  scatter_l0_kernel<<<gScatPos + gScatNeg, B, 0, stream>>>(
      pos_row, pos_col, e_pos, (const float4*)ego_pos, dis_pos, bop,
      neg_row, neg_col, e_neg, (const float4*)ego_neg, dis_neg, bon, gScatPos);
  gin_accum2_kernel<<<2 * gVec, B, 0, stream>>>(
      (const float4*)bop, (const float4*)bon,
      (float4*)ego_pos, (float4*)ego_neg,
      dis_pos, dis_neg, eps, 0,
      (float4*)out, (float4*)(out + per), gVec);

  // ---- Layer 1: both paths on pos graph (fused dual-path scatter) ----
  hipMemsetAsync(bop, 0, 2 * per * sizeof(float), stream);
  scatter2_kernel<<<gScatPos, B, 0, stream>>>(
      pos_row, pos_col, e_pos,
      (const float4*)ego_pos, (const float4*)ego_neg,
      dis_pos, bop, bon);
  gin_accum2_kernel<<<2 * gVec, B, 0, stream>>>(
      (const float4*)bop, (const float4*)bon,
      (float4*)ego_pos, (float4*)ego_neg,
      dis_pos, dis_pos, eps, 1,
      (float4*)out, (float4*)(out + per), gVec);
}